// NTMWriteHead_84799834292555
// MI455X (gfx1250) — compile-verified
//
#include <hip/hip_runtime.h>
#include <math.h>

#define B_   256
#define N_   2048
#define M_   64
#define C_   512
#define OUT_ 198
#define LDO  208   // padded leading dim for o

typedef float v2f __attribute__((ext_vector_type(2)));
typedef float v8f __attribute__((ext_vector_type(8)));

// ---------------------------------------------------------------------------
// Kernel 1: o = h @ W.T + b   (256x198x512 fp32 GEMM via V_WMMA_F32_16X16X4_F32)
// grid (16 m-tiles, 13 n-tiles), 1 wave per block.
// ---------------------------------------------------------------------------
__global__ void __launch_bounds__(32) ctrl_gemm(const float* __restrict__ h,
                                                const float* __restrict__ W,
                                                const float* __restrict__ bias,
                                                float* __restrict__ o) {
    const int m0   = blockIdx.x * 16;
    const int n0   = blockIdx.y * 16;
    const int lane = threadIdx.x;          // wave32
    const int lr   = lane & 15;
    const int kh   = (lane >> 4) << 1;     // lanes 16-31 hold K = k0+2, k0+3

    const int  nIdx = n0 + lr;
    const bool nOK  = nIdx < OUT_;

    const float* hrow = h + (size_t)(m0 + lr) * C_ + kh;           // A: rows m0+lr
    const float* wrow = W + (size_t)(nOK ? nIdx : 0) * C_ + kh;    // B: W[n][k]

    v8f acc = {};
    for (int k0 = 0; k0 < C_; k0 += 4) {
        v2f a, b;
        a.x = hrow[k0];
        a.y = hrow[k0 + 1];
        float bx = wrow[k0];
        float by = wrow[k0 + 1];
        b.x = nOK ? bx : 0.0f;
        b.y = nOK ? by : 0.0f;
        acc = __builtin_amdgcn_wmma_f32_16x16x4_f32(
            /*neg_a=*/false, a, /*neg_b=*/false, b,
            /*c_mod=*/(short)0, acc, /*reuse_a=*/false, /*reuse_b=*/false);
    }

    // C/D layout: VGPR r -> M = r (lanes 0-15) or r+8 (lanes 16-31), N = lane&15
    const int mBase = m0 + ((lane >> 4) << 3);
    if (nOK) {
        const float bb = bias[nIdx];
#pragma unroll
        for (int r = 0; r < 8; ++r)
            o[(size_t)(mBase + r) * LDO + nIdx] = acc[r] + bb;
    }
}

// ---------------------------------------------------------------------------
// Kernel 2: split o, apply activations, compute ||k+eps||.
// 256 blocks (one per b) x 64 threads.
// scal[b*8+..] = {beta, g, gamma, s0, s1, s2, knorm, pad}
// ---------------------------------------------------------------------------
__device__ __forceinline__ float softplus_f(float x) {
    return fmaxf(x, 0.0f) + log1pf(expf(-fabsf(x)));
}
__device__ __forceinline__ float sigmoid_f(float x) {
    return 1.0f / (1.0f + expf(-x));
}

__global__ void __launch_bounds__(64) ctrl_act(const float* __restrict__ o,
                                               float* __restrict__ k_e,
                                               float* __restrict__ e_sig,
                                               float* __restrict__ a_out,
                                               float* __restrict__ scal) {
    const int b = blockIdx.x;
    const int t = threadIdx.x;
    const float* orow = o + (size_t)b * LDO;
    __shared__ float red[64];

    // splits: k [0,64) beta [64] g [65] s [66,69) gamma [69] e [70,134) a [134,198)
    float kv = orow[t] + 1e-16f;
    k_e[b * M_ + t]   = kv;
    e_sig[b * M_ + t] = sigmoid_f(orow[70 + t]);
    a_out[b * M_ + t] = orow[134 + t];

    red[t] = kv * kv;
    __syncthreads();
    for (int s = 32; s > 0; s >>= 1) {
        if (t < s) red[t] += red[t + s];
        __syncthreads();
    }
    if (t == 0) {
        float* sc = scal + b * 8;
        sc[0] = softplus_f(orow[64]);          // beta
        sc[1] = sigmoid_f(orow[65]);           // g
        sc[2] = 1.0f + softplus_f(orow[69]);   // gamma
        float s0 = orow[66], s1 = orow[67], s2 = orow[68];
        float mx = fmaxf(s0, fmaxf(s1, s2));
        float e0 = expf(s0 - mx), e1 = expf(s1 - mx), e2 = expf(s2 - mx);
        float inv = 1.0f / (e0 + e1 + e2);
        sc[3] = e0 * inv; sc[4] = e1 * inv; sc[5] = e2 * inv;
        sc[6] = sqrtf(red[0]);                 // ||k+eps||
    }
}

// ---------------------------------------------------------------------------
// Block reduction (256 threads = 8 wave32): shuffle within wave + LDS across.
// ---------------------------------------------------------------------------
template <bool MAXOP>
__device__ __forceinline__ float blockReduce(float v, volatile float* scratch) {
#pragma unroll
    for (int off = 16; off > 0; off >>= 1) {
        float o = __shfl_down(v, off, 32);
        v = MAXOP ? fmaxf(v, o) : (v + o);
    }
    const int wave = threadIdx.x >> 5;
    const int lane = threadIdx.x & 31;
    if (lane == 0) scratch[wave] = v;
    __syncthreads();
    if (wave == 0) {
        v = (lane < 8) ? scratch[lane] : (MAXOP ? -INFINITY : 0.0f);
#pragma unroll
        for (int off = 4; off > 0; off >>= 1) {
            float o = __shfl_down(v, off, 32);
            v = MAXOP ? fmaxf(v, o) : (v + o);
        }
        if (lane == 0) scratch[0] = v;
    }
    __syncthreads();
    float r = scratch[0];
    __syncthreads();   // scratch reused by later reductions
    return r;
}

// ---------------------------------------------------------------------------
// Kernel 3: fused content addressing + softmax + gate + shift + sharpen + write.
// One block per batch element b; 256 threads; each thread owns 8 rows of N.
// ---------------------------------------------------------------------------
__global__ void __launch_bounds__(256) addressing_write(
    const float* __restrict__ memory, const float* __restrict__ w_prev,
    const float* __restrict__ k_e, const float* __restrict__ e_sig,
    const float* __restrict__ a_in, const float* __restrict__ scal,
    float* __restrict__ w_out, float* __restrict__ mem_out) {

    const int b = blockIdx.x;
    const int t = threadIdx.x;

    __shared__ float lw[N_];              // 8 KB: logits -> w_g
    __shared__ float lk[M_], le[M_], la[M_];
    __shared__ float red[8];

    if (t < M_) {
        lk[t] = k_e[b * M_ + t];
        le[t] = e_sig[b * M_ + t];
        la[t] = a_in[b * M_ + t];
    }
    const float beta  = scal[b * 8 + 0];
    const float g     = scal[b * 8 + 1];
    const float gamma = scal[b * 8 + 2];
    const float s0    = scal[b * 8 + 3];
    const float s1    = scal[b * 8 + 4];
    const float s2    = scal[b * 8 + 5];
    const float knorm = scal[b * 8 + 6];
    __syncthreads();

    const float4* memb = reinterpret_cast<const float4*>(memory + (size_t)b * N_ * M_);

    // ---- phase 1: cosine-similarity logits ----
    float lmax = -INFINITY;
    for (int r = 0; r < 8; ++r) {
        const int n = t + r * 256;
        const float4* row = memb + n * 16;
        float dot = 0.0f, ss = 0.0f;
#pragma unroll
        for (int i = 0; i < 16; ++i) {
            float4 v = row[i];
            float x0 = v.x + 1e-16f, x1 = v.y + 1e-16f;
            float x2 = v.z + 1e-16f, x3 = v.w + 1e-16f;
            dot += x0 * lk[4*i] + x1 * lk[4*i+1] + x2 * lk[4*i+2] + x3 * lk[4*i+3];
            ss  += x0 * x0 + x1 * x1 + x2 * x2 + x3 * x3;
        }
        float denom = fmaxf(sqrtf(ss) * knorm, 1e-8f);
        float logit = beta * dot / denom;
        lw[n] = logit;
        lmax  = fmaxf(lmax, logit);
    }

    // ---- phase 2: softmax over N + interpolation gate ----
    float gmax = blockReduce<true>(lmax, red);
    float ex[8];
    float lsum = 0.0f;
#pragma unroll
    for (int r = 0; r < 8; ++r) {
        const int n = t + r * 256;
        ex[r] = expf(lw[n] - gmax);
        lsum += ex[r];
    }
    float gsum = blockReduce<false>(lsum, red);
    float inv  = 1.0f / gsum;
#pragma unroll
    for (int r = 0; r < 8; ++r) {
        const int n = t + r * 256;
        float wc = ex[r] * inv;
        lw[n] = g * wc + (1.0f - g) * w_prev[(size_t)b * N_ + n];   // w_g
    }
    __syncthreads();

    // ---- phase 3: 3-tap circular shift + sharpening ----
    float wsv[8];
    float ssum = 0.0f;
#pragma unroll
    for (int r = 0; r < 8; ++r) {
        const int n  = t + r * 256;
        const int nm = (n + N_ - 1) & (N_ - 1);
        const int np = (n + 1) & (N_ - 1);
        float wt = lw[nm] * s0 + lw[n] * s1 + lw[np] * s2;   // w_t >= 0
        float v  = powf(wt, gamma);
        wsv[r] = v;
        ssum  += v;
    }
    float gss  = blockReduce<false>(ssum, red);
    float winv = 1.0f / (gss + 1e-16f);

    // ---- phase 4: emit w and erase/add memory update ----
    float4* outb = reinterpret_cast<float4*>(mem_out + (size_t)b * N_ * M_);
    for (int r = 0; r < 8; ++r) {
        const int n  = t + r * 256;
        const float wv = wsv[r] * winv;
        w_out[(size_t)b * N_ + n] = wv;
        const float4* row  = memb + n * 16;
        float4*       orow = outb + n * 16;
#pragma unroll
        for (int i = 0; i < 16; ++i) {
            float4 v = row[i];
            float4 o4;
            o4.x = v.x * (1.0f - wv * le[4*i+0]) + wv * la[4*i+0];
            o4.y = v.y * (1.0f - wv * le[4*i+1]) + wv * la[4*i+1];
            o4.z = v.z * (1.0f - wv * le[4*i+2]) + wv * la[4*i+2];
            o4.w = v.w * (1.0f - wv * le[4*i+3]) + wv * la[4*i+3];
            orow[i] = o4;
        }
    }
}

// ---------------------------------------------------------------------------
extern "C" void kernel_launch(void* const* d_in, const int* in_sizes, int n_in,
                              void* d_out, int out_size, void* d_ws, size_t ws_size,
                              hipStream_t stream) {
    const float* h      = (const float*)d_in[0];   // [B,C]
    const float* w_prev = (const float*)d_in[1];   // [B,N]
    const float* memory = (const float*)d_in[2];   // [B,N,M]
    const float* W      = (const float*)d_in[3];   // [OUT,C]
    const float* bias   = (const float*)d_in[4];   // [OUT]

    float* w_out   = (float*)d_out;                // [B,N]
    float* mem_out = w_out + (size_t)B_ * N_;      // [B,N,M]

    // workspace layout (floats)
    float* ws    = (float*)d_ws;
    float* o     = ws;                             // B_*LDO      = 53248
    float* k_e   = o   + (size_t)B_ * LDO;         // B_*M_       = 16384
    float* e_sig = k_e + (size_t)B_ * M_;          // B_*M_
    float* a_buf = e_sig + (size_t)B_ * M_;        // B_*M_
    float* scal  = a_buf + (size_t)B_ * M_;        // B_*8

    ctrl_gemm<<<dim3(16, 13), 32, 0, stream>>>(h, W, bias, o);
    ctrl_act<<<B_, 64, 0, stream>>>(o, k_e, e_sig, a_buf, scal);
    addressing_write<<<B_, 256, 0, stream>>>(memory, w_prev, k_e, e_sig, a_buf,
                                             scal, w_out, mem_out);
}